// ScalarReLU_20572893348680
// MI455X (gfx1250) — compile-verified
//
#include <hip/hip_runtime.h>

// Elementwise ReLU on 32x4096x1024 fp32 (134,217,728 elems).
// Pure HBM stream: ~1.07 GB traffic -> ~46 us floor at 23.3 TB/s.
// Bulk kernel: branch-free, 4x clause-formed global_load_b128 NT ->
// 16x v_max_num_f32 (single max per element via inline asm, no
// canonicalize pair) -> 4x clause-formed global_store_b128 NT.

typedef __attribute__((ext_vector_type(4))) float f4;

__device__ __forceinline__ float relu1(float v) {
    float r;
    // Single v_max_num_f32 with inline-constant 0; avoids the extra
    // quieting max that fmaxf lowering inserts.
    asm("v_max_num_f32 %0, %1, 0" : "=v"(r) : "v"(v));
    return r;
}

__device__ __forceinline__ f4 relu4(f4 v) {
    f4 r;
    r.x = relu1(v.x);
    r.y = relu1(v.y);
    r.z = relu1(v.z);
    r.w = relu1(v.w);
    return r;
}

// Every access in-bounds by construction: grid = nfull4 / 1024 blocks,
// each block covers float4 indices [blockIdx*1024, blockIdx*1024 + 1024).
__global__ __launch_bounds__(256) void relu_bulk_kernel(const f4* __restrict__ x,
                                                        f4* __restrict__ y) {
    const unsigned base = blockIdx.x * 1024u + threadIdx.x;  // float4 index

    f4 v0 = __builtin_nontemporal_load(&x[base]);
    f4 v1 = __builtin_nontemporal_load(&x[base + 256u]);
    f4 v2 = __builtin_nontemporal_load(&x[base + 512u]);
    f4 v3 = __builtin_nontemporal_load(&x[base + 768u]);

    __builtin_nontemporal_store(relu4(v0), &y[base]);
    __builtin_nontemporal_store(relu4(v1), &y[base + 256u]);
    __builtin_nontemporal_store(relu4(v2), &y[base + 512u]);
    __builtin_nontemporal_store(relu4(v3), &y[base + 768u]);
}

// Handles everything past the full-block region (partial float4 chunk and
// scalar remainder). Empty for the reference shape.
__global__ void relu_tail_kernel(const float* __restrict__ x,
                                 float* __restrict__ y,
                                 long long start, long long n) {
    long long i = start + (long long)blockIdx.x * blockDim.x + threadIdx.x;
    if (i < n) {
        y[i] = relu1(x[i]);
    }
}

extern "C" void kernel_launch(void* const* d_in, const int* in_sizes, int n_in,
                              void* d_out, int out_size, void* d_ws, size_t ws_size,
                              hipStream_t stream) {
    (void)n_in; (void)d_ws; (void)ws_size; (void)out_size;

    const float* x = (const float*)d_in[0];
    float* y = (float*)d_out;

    const long long n = (long long)in_sizes[0];       // elements
    const long long n4 = n / 4;                       // full float4s
    const long long per_block = 1024;                 // float4s per block
    const long long full_blocks = n4 / per_block;     // branch-free region
    const long long bulk_elems = full_blocks * per_block * 4;

    if (full_blocks > 0) {
        relu_bulk_kernel<<<(unsigned)full_blocks, 256, 0, stream>>>(
            (const f4*)x, (f4*)y);
    }

    if (bulk_elems < n) {
        const long long tail = n - bulk_elems;
        const int block = 256;
        const long long grid = (tail + block - 1) / block;
        relu_tail_kernel<<<(unsigned)grid, block, 0, stream>>>(x, y, bulk_elems, n);
    }
}